// Spatial_Net1_52991306498332
// MI455X (gfx1250) — compile-verified
//
#include <hip/hip_runtime.h>

typedef __attribute__((ext_vector_type(2))) float v2f;
typedef __attribute__((ext_vector_type(8))) float v8f;

// Problem constants (from reference)
#define N1n   85
#define F1n   256
#define O1n   200
#define N2n   5625
#define F2n   128
#define O2n   2
#define E1n   1360
#define E2n   180000
#define ROWSn 250
#define C1n   68      // 85*200  = 250*68
#define C2n   45      // 5625*2  = 250*45
#define CCn   113
#define OFn   5

// Padded staging buffer for GEMM1 output: 6x13 tiles of 16x16 -> 96 x 208.
// Lets the WMMA epilogue store unconditionally (no EXEC masking).
#define H1R   96
#define H1S   208

// ---------------------------------------------------------------------------
// Kernel 1: init aggregation buffers with bias, degrees with 1.0 (self loop)
// ---------------------------------------------------------------------------
__global__ void init_kernel(float* __restrict__ agg1, float* __restrict__ agg2,
                            float* __restrict__ deg1, float* __restrict__ deg2,
                            const float* __restrict__ b1, const float* __restrict__ b2) {
  int i = blockIdx.x * blockDim.x + threadIdx.x;
  if (i < N1n * O1n) agg1[i] = b1[i % O1n];
  if (i < N2n * O2n) agg2[i] = b2[i % O2n];
  if (i < N1n)       deg1[i] = 1.0f;   // self loop contributes 1 to degree
  if (i < N2n)       deg2[i] = 1.0f;
}

// ---------------------------------------------------------------------------
// Kernel 2: GEMM1  h1[85,200] = x1[85,256] @ W1[256,200]  via fp32 WMMA
// One wave (32 lanes) per 16x16 output tile, K stepped by 4.
// A 16x4 layout : lanes 0-15 -> K={k,k+1}, lanes 16-31 -> K={k+2,k+3}, M=lane&15
// B 4x16 layout : mirrored, N=lane&15
// C/D 16x16 f32 : VGPR i -> M = i (+8 for upper half-wave), N = lane&15
//
// Loads:  out-of-range rows of A / cols of B only feed rows/cols of D that
//         never reach the valid region, so addresses are CLAMPED (memory
//         safety only) and loads are unconditional.
// Stores: the staging buffer is padded to 96x208, so all 8 accumulator
//         elements store unconditionally.
// ---------------------------------------------------------------------------
__global__ void gemm1_wmma_kernel(const float* __restrict__ x1,
                                  const float* __restrict__ W1,
                                  float* __restrict__ h1p) {
  const int lane = threadIdx.x & 31;
  const int wave = blockIdx.x * (blockDim.x >> 5) + (threadIdx.x >> 5);
  const int MT = (N1n + 15) / 16;   // 6
  const int NT = (O1n + 15) / 16;   // 13
  if (wave >= MT * NT) return;      // uniform per-wave exit

  const int mt = wave / NT;
  const int nt = wave % NT;
  const int half = lane >> 4;       // 0 = lanes 0-15, 1 = lanes 16-31
  const int l    = lane & 15;
  const int row  = mt * 16 + l;     // A row this lane feeds
  const int col  = nt * 16 + l;     // B/D column this lane feeds

  // Clamped (memory-safe) load addresses.
  const int rowc = min(row, N1n - 1);
  const int colc = min(col, O1n - 1);

  const float* __restrict__ aptr = x1 + rowc * F1n + half * 2;  // +k0 each iter
  const float* __restrict__ bptr = W1 + half * 2 * O1n + colc;  // +k0*O1n each iter

  v8f acc = {};
#pragma unroll 4
  for (int k0 = 0; k0 < F1n; k0 += 4) {
    v2f a;
    a.x = aptr[k0];
    a.y = aptr[k0 + 1];
    v2f b;
    b.x = bptr[k0 * O1n];
    b.y = bptr[k0 * O1n + O1n];
    acc = __builtin_amdgcn_wmma_f32_16x16x4_f32(
        /*neg_a=*/false, a, /*neg_b=*/false, b,
        /*c_mod=*/(short)0, acc, /*reuse_a=*/false, /*reuse_b=*/false);
  }

  // Unconditional store into padded 96x208 staging buffer.
  float* __restrict__ dptr = h1p + (mt * 16 + half * 8) * H1S + nt * 16 + l;
#pragma unroll
  for (int i = 0; i < 8; ++i) {
    dptr[i * H1S] = acc[i];
  }
}

// ---------------------------------------------------------------------------
// Kernel 3: GEMM2  h2[5625,2] = x2[5625,128] @ W2[128,2]   (N=2 -> VALU dot)
// ---------------------------------------------------------------------------
__global__ void gemm2_kernel(const float* __restrict__ x2,
                             const float* __restrict__ W2,
                             float* __restrict__ h2) {
  const int r = blockIdx.x * blockDim.x + threadIdx.x;
  if (r >= N2n) return;
  const float* xr = x2 + r * F2n;
  float a0 = 0.0f, a1 = 0.0f;
#pragma unroll 4
  for (int k = 0; k < F2n; ++k) {
    const float xv = xr[k];
    a0 = fmaf(xv, W2[k * O2n + 0], a0);
    a1 = fmaf(xv, W2[k * O2n + 1], a1);
  }
  h2[r * O2n + 0] = a0;
  h2[r * O2n + 1] = a1;
}

// ---------------------------------------------------------------------------
// Kernel 4: degree accumulation over dst of each edge (self loops pre-counted)
// ---------------------------------------------------------------------------
__global__ void degree_kernel(const int* __restrict__ ei1, const int* __restrict__ ei2,
                              float* __restrict__ deg1, float* __restrict__ deg2) {
  const int total = E1n + E2n;
  for (int i = blockIdx.x * blockDim.x + threadIdx.x; i < total;
       i += gridDim.x * blockDim.x) {
    if (i < E1n) {
      atomicAdd(&deg1[ei1[E1n + i]], 1.0f);       // edge_index1[1][i]
    } else {
      const int e = i - E1n;
      atomicAdd(&deg2[ei2[E2n + e]], 1.0f);       // edge_index2[1][e]
    }
  }
}

// ---------------------------------------------------------------------------
// Kernel 5: deg -> deg^{-1/2} in place (deg >= 1 always due to self loop)
// ---------------------------------------------------------------------------
__global__ void dinv_kernel(float* __restrict__ deg1, float* __restrict__ deg2) {
  const int i = blockIdx.x * blockDim.x + threadIdx.x;
  if (i < N1n) deg1[i] = rsqrtf(deg1[i]);
  if (i < N2n) deg2[i] = rsqrtf(deg2[i]);
}

// ---------------------------------------------------------------------------
// Kernel 6/7: normalized scatter-add  agg[dst,f] += h[src,f]*dinv[src]*dinv[dst]
// Edges e >= E are the N self loops (src = dst = e - E).
// h1 is read from the padded (stride H1S) staging buffer.
// ---------------------------------------------------------------------------
__global__ void scatter1_kernel(const float* __restrict__ h1p, const int* __restrict__ ei1,
                                const float* __restrict__ dinv1, float* __restrict__ agg1) {
  const int total = (E1n + N1n) * O1n;
  for (int idx = blockIdx.x * blockDim.x + threadIdx.x; idx < total;
       idx += gridDim.x * blockDim.x) {
    const int e = idx / O1n;
    const int f = idx % O1n;
    int s, d;
    if (e < E1n) { s = ei1[e]; d = ei1[E1n + e]; }
    else         { s = d = e - E1n; }
    const float norm = dinv1[s] * dinv1[d];
    atomicAdd(&agg1[d * O1n + f], h1p[s * H1S + f] * norm);
  }
}

__global__ void scatter2_kernel(const float* __restrict__ h2, const int* __restrict__ ei2,
                                const float* __restrict__ dinv2, float* __restrict__ agg2) {
  const int total = (E2n + N2n) * O2n;
  for (int idx = blockIdx.x * blockDim.x + threadIdx.x; idx < total;
       idx += gridDim.x * blockDim.x) {
    const int e = idx / O2n;
    const int f = idx % O2n;
    int s, d;
    if (e < E2n) { s = ei2[e]; d = ei2[E2n + e]; }
    else         { s = d = e - E2n; }
    const float norm = dinv2[s] * dinv2[d];
    atomicAdd(&agg2[d * O2n + f], h2[s * O2n + f] * norm);
  }
}

// ---------------------------------------------------------------------------
// Kernel 8: concat([250,68],[250,45]) -> ReLU -> @ Wf[113,5] + bf
// ---------------------------------------------------------------------------
__global__ void final_kernel(const float* __restrict__ agg1, const float* __restrict__ agg2,
                             const float* __restrict__ Wf, const float* __restrict__ bf,
                             float* __restrict__ out) {
  const int idx = blockIdx.x * blockDim.x + threadIdx.x;
  if (idx >= ROWSn * OFn) return;
  const int r = idx / OFn;
  const int j = idx % OFn;
  float acc = bf[j];
#pragma unroll 1
  for (int c = 0; c < CCn; ++c) {
    float v = (c < C1n) ? agg1[r * C1n + c] : agg2[r * C2n + (c - C1n)];
    v = fmaxf(v, 0.0f);
    acc = fmaf(v, Wf[c * OFn + j], acc);
  }
  out[idx] = acc;
}

// ---------------------------------------------------------------------------
// Host side
// ---------------------------------------------------------------------------
extern "C" void kernel_launch(void* const* d_in, const int* in_sizes, int n_in,
                              void* d_out, int out_size, void* d_ws, size_t ws_size,
                              hipStream_t stream) {
  (void)in_sizes; (void)n_in; (void)out_size; (void)ws_size;

  const float* x1  = (const float*)d_in[0];
  const float* x2  = (const float*)d_in[1];
  const float* W1  = (const float*)d_in[2];
  const float* b1  = (const float*)d_in[3];
  const float* W2  = (const float*)d_in[4];
  const float* b2  = (const float*)d_in[5];
  const float* Wf  = (const float*)d_in[6];
  const float* bf  = (const float*)d_in[7];
  const int*   ei1 = (const int*)d_in[8];
  const int*   ei2 = (const int*)d_in[9];
  float* out = (float*)d_out;

  // Workspace layout (floats)
  float* ws   = (float*)d_ws;
  float* h1p  = ws;                       // 96*208 = 19968 (padded GEMM1 out)
  float* h2   = h1p + H1R * H1S;          // 11250
  float* agg1 = h2 + N2n * O2n;           // 17000
  float* agg2 = agg1 + N1n * O1n;         // 11250
  float* deg1 = agg2 + N2n * O2n;         // 85   (becomes dinv1)
  float* deg2 = deg1 + N1n;               // 5625 (becomes dinv2)

  // 1. init biases + self-loop degrees
  init_kernel<<<(N1n * O1n + 255) / 256, 256, 0, stream>>>(agg1, agg2, deg1, deg2, b1, b2);

  // 2. GEMM1 via fp32 WMMA: 78 tiles, 4 waves/block (wave32)
  gemm1_wmma_kernel<<<20, 128, 0, stream>>>(x1, W1, h1p);

  // 3. GEMM2 (N=2, VALU)
  gemm2_kernel<<<(N2n + 255) / 256, 256, 0, stream>>>(x2, W2, h2);

  // 4. degrees
  degree_kernel<<<((E1n + E2n) + 255) / 256, 256, 0, stream>>>(ei1, ei2, deg1, deg2);

  // 5. dinv
  dinv_kernel<<<(N2n + 255) / 256, 256, 0, stream>>>(deg1, deg2);

  // 6/7. normalized scatter aggregation
  scatter1_kernel<<<((E1n + N1n) * O1n + 255) / 256, 256, 0, stream>>>(h1p, ei1, deg1, agg1);
  scatter2_kernel<<<((E2n + N2n) * O2n + 255) / 256, 256, 0, stream>>>(h2, ei2, deg2, agg2);

  // 8. concat + ReLU + final linear
  final_kernel<<<(ROWSn * OFn + 255) / 256, 256, 0, stream>>>(agg1, agg2, Wf, bf, out);
}